// DecoderReal_90477781057695
// MI455X (gfx1250) — compile-verified
//
#include <hip/hip_runtime.h>
#include <math.h>

typedef __attribute__((ext_vector_type(2))) float v2f;
typedef __attribute__((ext_vector_type(8))) float v8f;

#define LATENT  32
#define ACTION  8
#define HIDDEN  64
#define OBS     20
#define TMAX    30
#define BATCH   16384
#define NSUB    8

// Hardware tanh (TRANS32 op on gfx1250).
__device__ __forceinline__ float fast_tanh(float x) {
#if __has_builtin(__builtin_amdgcn_tanhf)
    return __builtin_amdgcn_tanhf(x);
#else
    float r;
    // v_nop covers the 1-op TRANS result hazard the compiler can't see through asm.
    asm volatile("v_tanh_f32 %0, %1\n\tv_nop" : "=v"(r) : "v"(x));
    return r;
#endif
}

// ---------------------------------------------------------------------------
// Kernel 0: cum_a[t] = sum_{u<=t} a[u]   a: [30, B, 8]
// ---------------------------------------------------------------------------
__global__ __launch_bounds__(256) void cumsum_kernel(const float* __restrict__ a,
                                                     float* __restrict__ cum) {
    int i = blockIdx.x * 256 + threadIdx.x;
    if (i >= BATCH * ACTION) return;
    float acc = 0.0f;
    for (int t = 0; t < TMAX; ++t) {
        acc += a[(size_t)t * BATCH * ACTION + i];
        cum[(size_t)t * BATCH * ACTION + i] = acc;
    }
}

// ---------------------------------------------------------------------------
// One dynamics evaluation: dy = tanh( tanh([y,dose(t)] W1 + b1) W2 + b2 )
// Fragment layouts (V_WMMA_F32_16X16X4_F32):
//   A (16x4):  lane m / m+16 holds X[m][kb*4 + 2*cidx + {0,1}]   (cidx = lane>=16)
//   B (4x16):  lane n / n+16 holds W[kb*4 + 2*cidx + {0,1}][ntile*16+n]
//   C/D (16x16): vgpr j, lanes 0-15 -> M=j, lanes 16-31 -> M=j+8, N=lane&15
// Weights are pre-packed in LDS so each B fragment is one ds_load_b64.
// ---------------------------------------------------------------------------
__device__ __forceinline__ void dyn_eval(
    const v2f y[8], float t, int row, int cidx, int m, int n, int mrow,
    const float* __restrict__ cum_a,
    const v2f* pW1, const float* sb1,
    const v2f* pW2, const float* sb2,
    float* st, v2f dy[8])
{
    // --- dose(t) fragments (A-layout, 2 K-blocks) -------------------------
    v2f dose[2];
    int idx = (int)truncf(t);
    if (idx < 0) idx += TMAX;               // python negative-index wrap (t == -1)
    if (idx >= TMAX) {
        dose[0].x = 0.0f; dose[0].y = 0.0f;
        dose[1].x = 0.0f; dose[1].y = 0.0f;
    } else {
        const float* dp = cum_a + (size_t)idx * BATCH * ACTION + (size_t)row * ACTION + 2 * cidx;
        dose[0] = *(const v2f*)(dp);
        dose[1] = *(const v2f*)(dp + 4);
    }

    // --- layer 1: [16x40] @ [40x64] + b1 ; 4 independent accumulator chains
    v8f acc1[4];
#pragma unroll
    for (int nt = 0; nt < 4; ++nt) {
        float bias = sb1[nt * 16 + n];
#pragma unroll
        for (int j = 0; j < 8; ++j) acc1[nt][j] = bias;
    }
    asm volatile("" ::: "memory");
#pragma unroll
    for (int kb = 0; kb < 10; ++kb) {
        v2f af = (kb < 8) ? y[kb] : dose[kb - 8];
#pragma unroll
        for (int nt = 0; nt < 4; ++nt) {
            v2f bf = pW1[(kb * 2 + cidx) * HIDDEN + nt * 16 + n];
            acc1[nt] = __builtin_amdgcn_wmma_f32_16x16x4_f32(
                false, af, false, bf, (short)0, acc1[nt], false, false);
        }
    }
#pragma unroll
    for (int nt = 0; nt < 4; ++nt)
#pragma unroll
        for (int j = 0; j < 8; ++j)
            st[(j + mrow) * HIDDEN + nt * 16 + n] = fast_tanh(acc1[nt][j]);
    asm volatile("s_wait_dscnt 0" ::: "memory");   // cross-lane staging visible

    // --- reload h1 as A-layout fragments (one b64 each) -------------------
    v2f ha[16];
#pragma unroll
    for (int kb = 0; kb < 16; ++kb)
        ha[kb] = *(const v2f*)(st + m * HIDDEN + kb * 4 + 2 * cidx);
    asm volatile("" ::: "memory");

    // --- layer 2: [16x64] @ [64x32] + b2 ; 2 independent accumulator chains
    v8f acc2[2];
#pragma unroll
    for (int nt = 0; nt < 2; ++nt) {
        float bias = sb2[nt * 16 + n];
#pragma unroll
        for (int j = 0; j < 8; ++j) acc2[nt][j] = bias;
    }
#pragma unroll
    for (int kb = 0; kb < 16; ++kb) {
#pragma unroll
        for (int nt = 0; nt < 2; ++nt) {
            v2f bf = pW2[(kb * 2 + cidx) * LATENT + nt * 16 + n];
            acc2[nt] = __builtin_amdgcn_wmma_f32_16x16x4_f32(
                false, ha[kb], false, bf, (short)0, acc2[nt], false, false);
        }
    }
#pragma unroll
    for (int nt = 0; nt < 2; ++nt)
#pragma unroll
        for (int j = 0; j < 8; ++j)
            st[(j + mrow) * LATENT + nt * 16 + n] = fast_tanh(acc2[nt][j]);
    asm volatile("s_wait_dscnt 0" ::: "memory");

    // --- read dy back as A-layout fragments -------------------------------
#pragma unroll
    for (int kb = 0; kb < 8; ++kb)
        dy[kb] = *(const v2f*)(st + m * LATENT + kb * 4 + 2 * cidx);
    asm volatile("" ::: "memory");
}

// ---------------------------------------------------------------------------
// Kernel 1: fixed-step RK4 (NSUB substeps / unit interval); each wave32 owns
// a 16-row batch tile for the entire trajectory.
// ---------------------------------------------------------------------------
__global__ __launch_bounds__(256) void ode_kernel(
    const float* __restrict__ init,   // [B, 32]
    const float* __restrict__ cum_a,  // [30, B, 8]
    const float* __restrict__ W1, const float* __restrict__ b1,
    const float* __restrict__ W2, const float* __restrict__ b2,
    float* __restrict__ h_out)        // [31, B, 32]
{
    // Fragment-packed weights: pW1[(kb*2+c)*64 + nn] = {W1[kb*4+2c][nn], W1[kb*4+2c+1][nn]}
    __shared__ v2f   pW1[20 * HIDDEN];          // 10 K-blocks x 2 halves x 64 cols
    __shared__ v2f   pW2[32 * LATENT];          // 16 K-blocks x 2 halves x 32 cols
    __shared__ float sb1[HIDDEN];
    __shared__ float sb2[LATENT];
    __shared__ float stage[8][16 * HIDDEN];     // per-wave D<->A relayout buffer

    int tid = threadIdx.x;
    for (int i = tid; i < 20 * HIDDEN; i += 256) {
        int slot = i / HIDDEN;                  // kb*2 + c
        int nn   = i % HIDDEN;
        int kk   = (slot >> 1) * 4 + (slot & 1) * 2;
        v2f w; w.x = W1[kk * HIDDEN + nn]; w.y = W1[(kk + 1) * HIDDEN + nn];
        pW1[i] = w;
    }
    for (int i = tid; i < 32 * LATENT; i += 256) {
        int slot = i / LATENT;
        int nn   = i % LATENT;
        int kk   = (slot >> 1) * 4 + (slot & 1) * 2;
        v2f w; w.x = W2[kk * LATENT + nn]; w.y = W2[(kk + 1) * LATENT + nn];
        pW2[i] = w;
    }
    if (tid < HIDDEN) sb1[tid] = b1[tid];
    if (tid < LATENT) sb2[tid] = b2[tid];
    __syncthreads();

    const int wave = tid >> 5;
    const int lane = tid & 31;
    const int m    = lane & 15;
    const int n    = m;
    const int cidx = (lane < 16) ? 0 : 1;       // which K-half of each 4-block
    const int mrow = (lane < 16) ? 0 : 8;
    const int row  = (blockIdx.x * 8 + wave) * 16 + m;
    float* st = &stage[wave][0];

    // load y0 (A-layout) and emit h[0] = y(t = -1)
    v2f y[8];
#pragma unroll
    for (int kb = 0; kb < 8; ++kb) {
        y[kb] = *(const v2f*)(init + (size_t)row * LATENT + kb * 4 + 2 * cidx);
        *(v2f*)(h_out + (size_t)row * LATENT + kb * 4 + 2 * cidx) = y[kb];
    }

    const float hh = 1.0f / (float)NSUB;
    for (int ti = 0; ti < TMAX; ++ti) {            // interval [ti-1, ti]
        float t0 = -1.0f + (float)ti;
        for (int s = 0; s < NSUB; ++s) {
            float ts = t0 + (float)s * hh;
            v2f k[8], yt[8], acc[8];

            dyn_eval(y, ts, row, cidx, m, n, mrow, cum_a, pW1, sb1, pW2, sb2, st, k);
#pragma unroll
            for (int kb = 0; kb < 8; ++kb) { acc[kb] = k[kb]; yt[kb] = y[kb] + 0.5f * hh * k[kb]; }

            dyn_eval(yt, ts + 0.5f * hh, row, cidx, m, n, mrow, cum_a, pW1, sb1, pW2, sb2, st, k);
#pragma unroll
            for (int kb = 0; kb < 8; ++kb) { acc[kb] += 2.0f * k[kb]; yt[kb] = y[kb] + 0.5f * hh * k[kb]; }

            dyn_eval(yt, ts + 0.5f * hh, row, cidx, m, n, mrow, cum_a, pW1, sb1, pW2, sb2, st, k);
#pragma unroll
            for (int kb = 0; kb < 8; ++kb) { acc[kb] += 2.0f * k[kb]; yt[kb] = y[kb] + hh * k[kb]; }

            dyn_eval(yt, ts + hh, row, cidx, m, n, mrow, cum_a, pW1, sb1, pW2, sb2, st, k);
#pragma unroll
            for (int kb = 0; kb < 8; ++kb) y[kb] += (hh / 6.0f) * (acc[kb] + k[kb]);
        }
        // h[ti+1] = y(t = ti)
#pragma unroll
        for (int kb = 0; kb < 8; ++kb)
            *(v2f*)(h_out + ((size_t)(ti + 1) * BATCH + row) * LATENT + kb * 4 + 2 * cidx) = y[kb];
    }
}

// ---------------------------------------------------------------------------
// Kernel 2: x_hat = (elu(h Wo1 + bo1) Wo2 + bo2)[1:]
// ---------------------------------------------------------------------------
__global__ __launch_bounds__(256) void decoder_kernel(
    const float* __restrict__ h,      // [31, B, 32]
    const float* __restrict__ Wo1, const float* __restrict__ bo1,
    const float* __restrict__ Wo2, const float* __restrict__ bo2,
    float* __restrict__ xhat)         // [30, B, 20]
{
    __shared__ float sWo1[LATENT * (LATENT + 1)];
    __shared__ float sbo1[LATENT + 1];
    __shared__ float sWo2[(LATENT + 1) * OBS];
    __shared__ float sbo2[OBS];
    int tid = threadIdx.x;
    for (int i = tid; i < LATENT * (LATENT + 1); i += 256) sWo1[i] = Wo1[i];
    for (int i = tid; i < (LATENT + 1) * OBS; i += 256)    sWo2[i] = Wo2[i];
    if (tid < LATENT + 1) sbo1[tid] = bo1[tid];
    if (tid < OBS)        sbo2[tid] = bo2[tid];
    __syncthreads();

    int g = blockIdx.x * 256 + tid;
    if (g >= TMAX * BATCH) return;
    int t = g / BATCH + 1;            // 1..30 (drop first time point)
    int b = g % BATCH;

    float yv[LATENT];
#pragma unroll
    for (int i = 0; i < LATENT; ++i)
        yv[i] = h[((size_t)t * BATCH + b) * LATENT + i];

    float z[LATENT + 1];
#pragma unroll 1
    for (int o = 0; o < LATENT + 1; ++o) {
        float a = sbo1[o];
#pragma unroll
        for (int i = 0; i < LATENT; ++i) a += yv[i] * sWo1[i * (LATENT + 1) + o];
        z[o] = (a > 0.0f) ? a : expm1f(a);        // elu, alpha = 1
    }
#pragma unroll 1
    for (int o = 0; o < OBS; ++o) {
        float a = sbo2[o];
#pragma unroll
        for (int i = 0; i < LATENT + 1; ++i) a += z[i] * sWo2[i * OBS + o];
        xhat[((size_t)(t - 1) * BATCH + b) * OBS + o] = a;
    }
}

// ---------------------------------------------------------------------------
extern "C" void kernel_launch(void* const* d_in, const int* in_sizes, int n_in,
                              void* d_out, int out_size, void* d_ws, size_t ws_size,
                              hipStream_t stream) {
    const float* init = (const float*)d_in[0];   // [B, 32]
    const float* a    = (const float*)d_in[1];   // [30, B, 8]
    // d_in[2] = s : unused by reference dynamics
    const float* W1   = (const float*)d_in[3];
    const float* b1   = (const float*)d_in[4];
    const float* W2   = (const float*)d_in[5];
    const float* b2   = (const float*)d_in[6];
    const float* Wo1  = (const float*)d_in[7];
    const float* bo1  = (const float*)d_in[8];
    const float* Wo2  = (const float*)d_in[9];
    const float* bo2  = (const float*)d_in[10];

    float* xhat = (float*)d_out;                       // [30, B, 20]
    float* h    = xhat + (size_t)TMAX * BATCH * OBS;   // [31, B, 32]
    float* cum  = (float*)d_ws;                        // [30, B, 8]

    cumsum_kernel<<<(BATCH * ACTION + 255) / 256, 256, 0, stream>>>(a, cum);
    ode_kernel<<<BATCH / 128, 256, 0, stream>>>(init, cum, W1, b1, W2, b2, h);
    decoder_kernel<<<(TMAX * BATCH + 255) / 256, 256, 0, stream>>>(h, Wo1, bo1, Wo2, bo2, xhat);
}